// SCIBERTCRF_12747462934910
// MI455X (gfx1250) — compile-verified
//
#include <hip/hip_runtime.h>
#include <hip/hip_bf16.h>

typedef __attribute__((ext_vector_type(2))) float v2f;
typedef __attribute__((ext_vector_type(8))) float v8f;

#define B_   64
#define S_   512
#define D_   768
#define NTOK (B_ * S_)

__device__ __forceinline__ float lse2(float x, float y) {
    float m = fmaxf(x, y);
    return m + __logf(__expf(x - m) + __expf(y - m));
}

// ---------------------------------------------------------------------------
// Kernel 1: masked emissions via V_WMMA_F32_16X16X4_F32.
// One wave = one 16-token tile. seq rows staged through LDS (stride-68 pad),
// W (2x768) cached in LDS, N-columns 2..15 of B fed zeros.
// Also emits mask / labels_m (float) directly into d_out.
// ---------------------------------------------------------------------------
__global__ __launch_bounds__(128) void scibert_emissions_kernel(
    const float* __restrict__ seq, const long long* __restrict__ ids,
    const long long* __restrict__ labels, const float* __restrict__ W,
    const float* __restrict__ bias, float* __restrict__ em,
    float* __restrict__ outLab, float* __restrict__ outMask)
{
    __shared__ float sW[2 * D_];        // 6 KB
    __shared__ float sA[4][16 * 68];    // 4 waves * 4.25 KB, padded vs bank conflicts
    __shared__ float sMask[4][16];

    const int tid  = threadIdx.x;
    const int wave = tid >> 5;
    const int lane = tid & 31;
    const int half = lane >> 4;   // lane group: K-subtile selector
    const int q    = lane & 15;   // M (for A) / N (for B) index

    for (int i = tid; i < 2 * D_; i += 128) sW[i] = W[i];

    const int tile = blockIdx.x * 4 + wave;   // 2048 tiles of 16 tokens
    const int tok0 = tile * 16;

    if (lane < 16) {
        int tok = tok0 + lane;
        long long id = ids[tok];
        long long lb = labels[tok];
        float mf = ((id != 0) && (id != 103) && (lb != 100)) ? 1.0f : 0.0f;
        sMask[wave][lane] = mf;
        outMask[tok] = mf;
        outLab[tok]  = mf * (float)lb;
    }
    __syncthreads();

    v8f acc = {};
    for (int c = 0; c < 12; ++c) {
        // Stage 16 tokens x 64 floats, masked, coalesced float4 per row-pair.
        #pragma unroll
        for (int j = 0; j < 8; ++j) {
            int r = 2 * j + half;
            const float4 v = *(const float4*)(seq + (size_t)(tok0 + r) * D_ + c * 64 + q * 4);
            float mf = sMask[wave][r];
            float4 w; w.x = v.x * mf; w.y = v.y * mf; w.z = v.z * mf; w.w = v.w * mf;
            *(float4*)&sA[wave][r * 68 + q * 4] = w;
        }
        if (c + 1 < 12)
            __builtin_prefetch(seq + (size_t)(tok0 + half) * D_ + (c + 1) * 64 + q * 4, 0, 3);
        __syncthreads();

        #pragma unroll
        for (int k = 0; k < 64; k += 4) {
            // A 16x4 f32: lanes0-15 K=k,k+1 ; lanes16-31 K=k+2,k+3 ; M = q
            v2f a, bb;
            a.x = sA[wave][q * 68 + k + 2 * half];
            a.y = sA[wave][q * 68 + k + 2 * half + 1];
            // B 4x16 f32: VGPR r holds rows K=r (lanes0-15) / K=r+2 (lanes16-31); N = q
            int kg = c * 64 + k + 2 * half;
            bb.x = (q < 2) ? sW[q * D_ + kg]     : 0.0f;
            bb.y = (q < 2) ? sW[q * D_ + kg + 1] : 0.0f;
            acc = __builtin_amdgcn_wmma_f32_16x16x4_f32(
                false, a, false, bb, (short)0, acc, false, false);
        }
        __syncthreads();
    }

    // C layout: VGPR r -> M = r (lanes0-15) / M = 8+r (lanes16-31), N = q.
    if (q < 2) {
        float bv = bias[q];
        #pragma unroll
        for (int r = 0; r < 8; ++r) {
            int tokm = tok0 + r + 8 * half;
            float e = acc[r] + bv;
            em[(size_t)tokm * 2 + q] = e > 0.0f ? e : 0.0f;
        }
    }
}

// ---------------------------------------------------------------------------
// Kernel 2: CRF token-mean log-likelihood + Viterbi decode. T=2, thread=batch.
// ---------------------------------------------------------------------------
__global__ __launch_bounds__(64) void scibert_crf_kernel(
    const float* __restrict__ em, const long long* __restrict__ ids,
    const long long* __restrict__ labels, const float* __restrict__ start,
    const float* __restrict__ endt, const float* __restrict__ trans,
    float* __restrict__ out, unsigned char* __restrict__ bp)
{
    __shared__ float red[64];
    const int b = threadIdx.x;
    const float tr00 = trans[0], tr01 = trans[1], tr10 = trans[2], tr11 = trans[3];
    const float en0 = endt[0], en1 = endt[1];
    const float* e = em + (size_t)b * S_ * 2;
    const long long* idp = ids + (size_t)b * S_;
    const long long* lbp = labels + (size_t)b * S_;

    auto tg = [&](int s) -> int {
        long long id = idp[s]; long long lb = lbp[s];
        return ((id != 0) && (id != 103) && (lb != 100)) ? (int)lb : 0;
    };

    float a0 = start[0] + e[0];
    float a1 = start[1] + e[1];
    float v0 = a0, v1 = a1;
    int prev = tg(0);
    float score = start[prev] + e[prev];

    for (int s = 1; s < S_; ++s) {
        float e0 = e[s * 2 + 0], e1 = e[s * 2 + 1];
        // forward (logsumexp over prev tag)
        float n0 = lse2(a0 + tr00, a1 + tr10) + e0;
        float n1 = lse2(a0 + tr01, a1 + tr11) + e1;
        a0 = n0; a1 = n1;
        // viterbi (argmax: first max on tie, matching jnp.argmax)
        float c00 = v0 + tr00, c10 = v1 + tr10;
        float c01 = v0 + tr01, c11 = v1 + tr11;
        bp[((size_t)b * S_ + s) * 2 + 0] = (unsigned char)((c10 > c00) ? 1 : 0);
        bp[((size_t)b * S_ + s) * 2 + 1] = (unsigned char)((c11 > c01) ? 1 : 0);
        v0 = fmaxf(c00, c10) + e0;
        v1 = fmaxf(c01, c11) + e1;
        // gold-path score
        int t = tg(s);
        score += ((prev == 0) ? (t == 0 ? tr00 : tr01) : (t == 0 ? tr10 : tr11))
               + e[s * 2 + t];
        prev = t;
    }
    score += (prev == 0) ? en0 : en1;
    float logZ = lse2(a0 + en0, a1 + en1);
    float llh = score - logZ;

    // backtrace
    float f0 = v0 + en0, f1 = v1 + en1;
    int cur = (f1 > f0) ? 1 : 0;
    float* tgt = out + 1;
    tgt[b * S_ + (S_ - 1)] = (float)cur;
    for (int s = S_ - 1; s >= 1; --s) {
        cur = bp[((size_t)b * S_ + s) * 2 + cur];
        tgt[b * S_ + (s - 1)] = (float)cur;
    }

    red[b] = llh;
    __syncthreads();
    if (b == 0) {
        float tsum = 0.0f;
        for (int i = 0; i < 64; ++i) tsum += red[i];
        out[0] = tsum / (float)(B_ * S_);
    }
}

// ---------------------------------------------------------------------------
// d_out layout (reference return order, all as f32):
//   [0]                      loss
//   [1            , 1+32768) target_emissions
//   [1+32768      , 1+65536) labels_m
//   [1+65536      , 1+98304) mask
// d_ws: em (256 KB f32) then backpointers (64 KB u8).
// ---------------------------------------------------------------------------
extern "C" void kernel_launch(void* const* d_in, const int* in_sizes, int n_in,
                              void* d_out, int out_size, void* d_ws, size_t ws_size,
                              hipStream_t stream)
{
    const float*     seq    = (const float*)d_in[0];
    const long long* ids    = (const long long*)d_in[1];
    const long long* labels = (const long long*)d_in[2];
    const float*     W      = (const float*)d_in[3];
    const float*     bias   = (const float*)d_in[4];
    const float*     start  = (const float*)d_in[5];
    const float*     endt   = (const float*)d_in[6];
    const float*     trans  = (const float*)d_in[7];

    float* out = (float*)d_out;
    float* em  = (float*)d_ws;
    unsigned char* bp = (unsigned char*)d_ws + (size_t)NTOK * 2 * sizeof(float);

    float* outLab  = out + 1 + NTOK;
    float* outMask = out + 1 + 2 * NTOK;

    scibert_emissions_kernel<<<dim3(NTOK / 64), dim3(128), 0, stream>>>(
        seq, ids, labels, W, bias, em, outLab, outMask);
    scibert_crf_kernel<<<dim3(1), dim3(64), 0, stream>>>(
        em, ids, labels, start, endt, trans, out, bp);
}